// VectorQuantizer_29446295781419
// MI455X (gfx1250) — compile-verified
//
#include <hip/hip_runtime.h>

typedef __attribute__((ext_vector_type(2))) float v2f;
typedef __attribute__((ext_vector_type(8))) float v8f;

namespace {
constexpr int kD      = 64;        // embedding dim
constexpr int kK      = 1024;      // number of codes
constexpr int kHW     = 1024;      // H*W
constexpr int kNRows  = 65536;     // B*H*W
constexpr int kQElems = 64 * 64 * 32 * 32;  // 4194304 = q_out element count
constexpr int kTiles  = kK / 16;   // 64 code tiles of 16 codes
constexpr int kKK     = kD / 4;    // 16 K-steps of 4 per tile (f32 WMMA K=4)

// workspace layout (in floats)
constexpr int WS_ENORM = 0;                 // [1024]  ||e_k||^2
constexpr int WS_BPACK = 1024;              // [65536] codebook in B-fragment layout
constexpr int WS_PART  = 1024 + 65536;      // [4096]  per-wave partial loss sums
}

// ---------------------------------------------------------------------------
// Prep: swizzle codebook into WMMA B-fragment order + code norms.
// B 4x16 f32 fragment: lane L holds column n=L&15, K elements {hi*2, hi*2+1}
// with hi = L>>4, for K-step kk covering dims 4*kk .. 4*kk+3.
// bpack[((t*16+kk)*32 + L)*2 + {0,1}] = emb[(t*16 + (L&15))*64 + 4*kk + hi*2 + {0,1}]
// ---------------------------------------------------------------------------
__global__ void vq_prep(const float* __restrict__ emb, float* ws) {
  const int tid = blockIdx.x * blockDim.x + threadIdx.x;   // 0..32767
  const int t  = tid >> 9;
  const int kk = (tid >> 5) & 15;
  const int ln = tid & 31;
  const int code = t * 16 + (ln & 15);
  const int d = kk * 4 + ((ln >> 4) << 1);
  const float* src = emb + code * kD + d;
  float* dst = ws + WS_BPACK + tid * 2;
  dst[0] = src[0];
  dst[1] = src[1];
  if (tid < kK) {
    const float* row = emb + tid * kD;
    float s = 0.0f;
    #pragma unroll
    for (int j = 0; j < kD; ++j) s += row[j] * row[j];
    ws[WS_ENORM + tid] = s;
  }
}

// ---------------------------------------------------------------------------
// Main: per-wave 16-row tile, sweep 64 code tiles with f32 WMMA, track argmin,
// gather winning code rows to q_out, accumulate squared error.
// ---------------------------------------------------------------------------
__global__ void __launch_bounds__(256) vq_main(const float* __restrict__ latents,
                                               const float* __restrict__ emb,
                                               float* ws,
                                               float* __restrict__ out) {
  __shared__ int kmin[8][16];

  const int lane = threadIdx.x & 31;
  const int wave = threadIdx.x >> 5;
  const int m    = lane & 15;       // row within tile / column within code tile
  const int hi   = lane >> 4;
  const int rowBase = blockIdx.x * 128 + wave * 16;
  const int b      = rowBase / kHW;       // 128 | 1024, never straddles a batch
  const int hwBase = rowBase % kHW;

  // A fragment (16x64 f32): lane holds row m, dims {4kk + hi*2, +1} per step.
  // NCHW layout => element (row, d) at (b*64 + d)*1024 + hw; loads coalesce
  // across the 16 consecutive hw values.
  v2f a[kKK];
  #pragma unroll
  for (int kk = 0; kk < kKK; ++kk) {
    const int d = kk * 4 + hi * 2;
    const size_t off = (size_t)(b * kD + d) * kHW + hwBase + m;
    v2f av;
    av[0] = latents[off];
    av[1] = latents[off + kHW];   // d+1 plane
    a[kk] = av;
  }

  const float* bpack = ws + WS_BPACK;
  float best[8];
  int   bidx[8];
  #pragma unroll
  for (int r = 0; r < 8; ++r) { best[r] = 3.4e38f; bidx[r] = 0; }

  #pragma unroll 1
  for (int t = 0; t < kTiles; ++t) {
    if (t + 1 < kTiles)
      __builtin_prefetch(bpack + (size_t)(t + 1) * 1024, 0, 1);
    const float* btile = bpack + (size_t)t * 1024;   // 16 kk-steps * 32 lanes * 2
    v8f c = {};
    #pragma unroll
    for (int kk = 0; kk < kKK; ++kk) {
      v2f bv = *(const v2f*)(btile + (kk * 32 + lane) * 2);
      c = __builtin_amdgcn_wmma_f32_16x16x4_f32(false, a[kk], false, bv,
                                                (short)0, c, false, false);
    }
    // score = ||e||^2 - 2*dot ; lane's column = code t*16 + m, rows r + hi*8
    const int code = t * 16 + m;
    const float en = ws[WS_ENORM + code];
    #pragma unroll
    for (int r = 0; r < 8; ++r) {
      const float score = en - 2.0f * c[r];
      if (score < best[r]) { best[r] = score; bidx[r] = code; }  // strict < keeps
    }                                                            // first (lowest) code
  }

  // Butterfly min across the 16 lanes sharing each row set; tie -> lower index.
  #pragma unroll
  for (int r = 0; r < 8; ++r) {
    float bv = best[r];
    int   bi = bidx[r];
    #pragma unroll
    for (int msk = 1; msk < 16; msk <<= 1) {
      const float ov = __shfl_xor(bv, msk, 32);
      const int   oi = __shfl_xor(bi, msk, 32);
      if (ov < bv || (ov == bv && oi < bi)) { bv = ov; bi = oi; }
    }
    if (m == 0) kmin[wave][hi * 8 + r] = bi;   // lanes 0 / 16 publish rows 0-7 / 8-15
  }
  __syncthreads();

  // encoding indices (stored as float in the float output buffer)
  if (hi == 0) out[kQElems + 2 + rowBase + m] = (float)kmin[wave][m];

  // Gather winning code row, write q_out (NCHW, coalesced), accumulate loss.
  const int km = kmin[wave][m];
  const float* erow = emb + km * kD;
  float lsum = 0.0f;
  #pragma unroll 4
  for (int dd = 0; dd < 32; ++dd) {
    const int d = dd * 2 + hi;   // hi=0 lanes: even dims; hi=1 lanes: odd dims
    const size_t off = (size_t)(b * kD + d) * kHW + hwBase + m;
    const float ev = erow[d];
    const float xv = latents[off];
    const float df = ev - xv;
    lsum += df * df;
    out[off] = ev;
  }
  // deterministic wave reduction
  #pragma unroll
  for (int msk = 1; msk < 32; msk <<= 1) lsum += __shfl_xor(lsum, msk, 32);
  if (lane == 0) ws[WS_PART + blockIdx.x * 8 + wave] = lsum;
}

// ---------------------------------------------------------------------------
// Finalize: deterministic tree-sum of 4096 partials -> both loss scalars.
// ---------------------------------------------------------------------------
__global__ void vq_finalize(const float* __restrict__ ws, float* __restrict__ out) {
  __shared__ float s[256];
  const int tid = threadIdx.x;
  float acc = 0.0f;
  #pragma unroll
  for (int j = 0; j < 16; ++j) acc += ws[WS_PART + tid * 16 + j];
  s[tid] = acc;
  __syncthreads();
  for (int stride = 128; stride > 0; stride >>= 1) {
    if (tid < stride) s[tid] += s[tid + stride];
    __syncthreads();
  }
  if (tid == 0) {
    const float mse = s[0] / (float)((long long)kNRows * kD);
    out[kQElems]     = mse;   // commitment_loss
    out[kQElems + 1] = mse;   // embedding_loss (same value)
  }
}

extern "C" void kernel_launch(void* const* d_in, const int* in_sizes, int n_in,
                              void* d_out, int out_size, void* d_ws, size_t ws_size,
                              hipStream_t stream) {
  (void)in_sizes; (void)n_in; (void)out_size; (void)ws_size;
  const float* latents = (const float*)d_in[0];
  const float* emb     = (const float*)d_in[1];
  float* out = (float*)d_out;
  float* ws  = (float*)d_ws;

  hipLaunchKernelGGL(vq_prep,     dim3(128), dim3(256), 0, stream, emb, ws);
  hipLaunchKernelGGL(vq_main,     dim3(512), dim3(256), 0, stream, latents, emb, ws, out);
  hipLaunchKernelGGL(vq_finalize, dim3(1),   dim3(256), 0, stream, ws, out);
}